// ResidualBlock_80745385165393
// MI455X (gfx1250) — compile-verified
//
#include <hip/hip_runtime.h>
#include <math.h>
#include <stdint.h>

// ---------------------------------------------------------------------------
// HeteroGraphConv (GraphConv norm='both' + BatchNorm, sum cross-etype agg,
// 2 layers, residual) for MI455X / gfx1250.
//
// Design (memory-bound workload, 23.3 TB/s HBM):
//  * edge scatter = atomicAdd stream (dominant cost, ~5.7 GB/layer)
//  * GEMM agg @ W via V_WMMA_F32_16X16X4_F32 (fp32-faithful; compute is far
//    from the matrix ceiling so no precision downcast)
//  * A tile staged HBM->LDS with GLOBAL_LOAD_ASYNC_TO_LDS_B128 (ASYNCcnt),
//    dst-side degree norm folded into the epilogue (row-scale commutes with
//    the K-reduction), BatchNorm fused into GEMM epilogues (stats pass +
//    normalize pass) to avoid materializing h (~410 MB saved per conv)
// ---------------------------------------------------------------------------

#define D      512
#define NA_    100000
#define NB_    100000
#define NG_    4096
#define BN_EPS 1e-5f

#define BM  16          // block tile rows (M)
#define BN  128         // block tile cols (N) = 8 waves x 16
#define LDA 516         // padded LDS row stride (floats): 516 % 64 = 4 banks skew

typedef __attribute__((ext_vector_type(2))) float v2f;
typedef __attribute__((ext_vector_type(8))) float v8f;

// ------------------------------- utilities --------------------------------

__global__ void k_zero(float* __restrict__ p, long n) {
  long i = (long)blockIdx.x * blockDim.x + threadIdx.x;
  long s = (long)gridDim.x * blockDim.x;
  for (; i < n; i += s) p[i] = 0.0f;
}

__global__ void k_copy(float* __restrict__ dst, const float* __restrict__ src, long n) {
  long i = (long)blockIdx.x * blockDim.x + threadIdx.x;
  long s = (long)gridDim.x * blockDim.x;
  for (; i < n; i += s) dst[i] = src[i];
}

// degree count (exact in fp32 up to 2^24 edges; we have <=2e5 per etype)
__global__ void k_deg(const int* __restrict__ src, const int* __restrict__ dst,
                      long ne, float* __restrict__ deg_o, float* __restrict__ deg_i) {
  long i = (long)blockIdx.x * blockDim.x + threadIdx.x;
  long s = (long)gridDim.x * blockDim.x;
  for (; i < ne; i += s) {
    atomicAdd(&deg_o[src[i]], 1.0f);
    atomicAdd(&deg_i[dst[i]], 1.0f);
  }
}

__global__ void k_rsqrt_deg(float* __restrict__ p, long n) {
  long i = (long)blockIdx.x * blockDim.x + threadIdx.x;
  long s = (long)gridDim.x * blockDim.x;
  for (; i < n; i += s) p[i] = rsqrtf(fmaxf(p[i], 1.0f));
}

// agg[dst] += x[src] * rsqrt(deg_out[src])   (consecutive threads -> same edge,
// consecutive feature cols -> coalesced x reads / agg atomics)
__global__ void k_scatter(const float* __restrict__ x, const int* __restrict__ src,
                          const int* __restrict__ dst, const float* __restrict__ rs_o,
                          float* __restrict__ agg, long ne) {
  long total = ne << 9;   // ne * 512
  long i = (long)blockIdx.x * blockDim.x + threadIdx.x;
  long s = (long)gridDim.x * blockDim.x;
  for (; i < total; i += s) {
    long e = i >> 9;
    int  j = (int)(i & (D - 1));
    int  sv = src[e];
    float v = x[(long)sv * D + j] * rs_o[sv];
    atomicAdd(&agg[(long)dst[e] * D + j], v);
  }
}

// mu / invstd from column sums
__global__ void k_bnstats(const float* __restrict__ colsum, const float* __restrict__ colsq,
                          float* __restrict__ mu, float* __restrict__ inv, float n) {
  int c = threadIdx.x;              // launched with D threads
  float m   = colsum[c] / n;
  float var = colsq[c] / n - m * m;
  mu[c]  = m;
  inv[c] = rsqrtf(var + BN_EPS);
}

// ------------------------- WMMA GEMM + fused BN ----------------------------
// h = rs_in[row] * (agg[row,:] @ W) + bias     (row-scale folded into epilogue)
// pass 1: accumulate per-column sum(h), sum(h^2)
// pass 2: out[row,col] += gamma*(h - mu)*inv + beta
__global__ __launch_bounds__(256)
void k_gemm_bn(const float* __restrict__ agg, const float* __restrict__ rs_in,
               const float* __restrict__ W,   const float* __restrict__ bias,
               float* __restrict__ colsum,    float* __restrict__ colsq,
               const float* __restrict__ mu,  const float* __restrict__ inv,
               const float* __restrict__ gamma, const float* __restrict__ beta,
               float* __restrict__ out, int pass) {
  __shared__ float As[BM * LDA];    // 33 KB of the 320 KB WGP LDS

  const int tid  = threadIdx.x;
  const int wave = tid >> 5;
  const int lane = tid & 31;
  const int row0 = blockIdx.x * BM;
  const int col0 = blockIdx.y * BN;

  // ---- async-copy A tile (16 x 512 f32) HBM -> LDS, no VGPR bounce ----
  // 2048 x 16B chunks over 256 threads; consecutive lanes -> consecutive 16B.
  // Flat shared pointer: addr[31:0] is the workgroup-relative LDS byte offset.
  const unsigned lds0 = (unsigned)(uintptr_t)(void*)As;
  for (int i = tid; i < BM * (D / 4); i += 256) {
    int r  = i >> 7;        // 128 float4 per row
    int c4 = i & 127;
    uint64_t g = (uint64_t)(uintptr_t)(agg + (long)(row0 + r) * D + (c4 << 2));
    unsigned l = lds0 + ((unsigned)(r * LDA + (c4 << 2)) << 2);
    asm volatile("global_load_async_to_lds_b128 %0, %1, off"
                 :: "v"(l), "v"(g) : "memory");
  }
  asm volatile("s_wait_asynccnt 0" ::: "memory");
  __syncthreads();

  // fragment coordinates per CDNA5 ISA layouts (wave32):
  //  A 16x4 f32 : lane&15 = M row, (lane>>4)*2 = K pair base
  //  B 4x16 f32 : lane&15 = N col, (lane>>4)*2 = K pair base
  //  C/D 16x16  : VGPR v -> row v + 8*(lane>>4), col = lane&15
  const int m    = lane & 15;
  const int koff = (lane >> 4) << 1;
  const int col  = col0 + (wave << 4) + (lane & 15);

  v8f acc = {0.f, 0.f, 0.f, 0.f, 0.f, 0.f, 0.f, 0.f};
  const float* As_row = As + m * LDA;
  const float* Wc     = W + col;

  #pragma unroll 4
  for (int k = 0; k < D; k += 4) {
    v2f a = *(const v2f*)(As_row + k + koff);          // ds_load_b64
    v2f b;
    b.x = Wc[(long)(k + koff) * D];                    // coalesced across lanes; W L2-resident
    b.y = Wc[(long)(k + koff + 1) * D];
    acc = __builtin_amdgcn_wmma_f32_16x16x4_f32(
        /*neg_a=*/false, a, /*neg_b=*/false, b,
        /*c_mod=*/(short)0, acc, /*reuse_a=*/false, /*reuse_b=*/false);
  }

  // epilogue: apply dst-side degree norm per-row + bias
  const float bv    = bias[col];
  const int   rbase = row0 + ((lane >> 4) << 3);
  float h[8];
  #pragma unroll
  for (int v = 0; v < 8; ++v)
    h[v] = rs_in[rbase + v] * acc[v] + bv;

  if (pass == 1) {
    float s = 0.f, sq = 0.f;
    #pragma unroll
    for (int v = 0; v < 8; ++v) { s += h[v]; sq += h[v] * h[v]; }
    // lanes L and L+16 hold the same column (rows 0-7 vs 8-15): fold halves
    s  += __shfl_xor(s, 16, 32);
    sq += __shfl_xor(sq, 16, 32);
    if (lane < 16) {
      atomicAdd(&colsum[col], s);
      atomicAdd(&colsq[col], sq);
    }
  } else {
    const float g = gamma[col], be = beta[col], mm = mu[col], iv = inv[col];
    #pragma unroll
    for (int v = 0; v < 8; ++v)
      out[(long)(rbase + v) * D + col] += (h[v] - mm) * iv * g + be;  // unique (row,col)
  }
}

// ------------------------------- launcher ----------------------------------

extern "C" void kernel_launch(void* const* d_in, const int* in_sizes, int n_in,
                              void* d_out, int out_size, void* d_ws, size_t ws_size,
                              hipStream_t stream) {
  (void)n_in; (void)out_size; (void)ws_size;

  const int NNODES[3] = {NA_, NB_, NG_};
  // ETYPES order (== alphabetical): a2a a2b a2g b2a b2b b2g g2a g2b g2g
  const int SRC_T[9] = {0, 0, 0, 1, 1, 1, 2, 2, 2};
  const int DST_T[9] = {0, 1, 2, 0, 1, 2, 0, 1, 2};

  const float* feat[3] = {(const float*)d_in[0], (const float*)d_in[1], (const float*)d_in[2]};
  const int* esrc[9]; const int* edst[9]; long ne[9];
  for (int t = 0; t < 9; ++t) {
    esrc[t] = (const int*)d_in[3 + 2 * t];
    edst[t] = (const int*)d_in[4 + 2 * t];
    ne[t]   = in_sizes[3 + 2 * t];
  }

  // params flattened as a JAX pytree: dict keys sorted -> '{et}_{l}' blocks
  // interleave layers (a2a_0, a2a_1, a2b_0, ...); inner keys sorted: W,b,beta,gamma
  const float *Wp[2][9], *bp[2][9], *betap[2][9], *gammap[2][9];
  for (int t = 0; t < 9; ++t)
    for (int l = 0; l < 2; ++l) {
      int base = 21 + (t * 2 + l) * 4;
      Wp[l][t]     = (const float*)d_in[base + 0];
      bp[l][t]     = (const float*)d_in[base + 1];
      betap[l][t]  = (const float*)d_in[base + 2];
      gammap[l][t] = (const float*)d_in[base + 3];
    }

  // ---- workspace carve-up (floats) ----
  float* ws = (float*)d_ws;
  long off = 0;
  float* out0[3];
  out0[0] = ws + off; off += (long)NA_ * D;
  out0[1] = ws + off; off += (long)NB_ * D;
  out0[2] = ws + off; off += (long)NG_ * D;
  float* agg = ws + off; off += (long)NA_ * D;      // max n_dst
  long rs_base = off;
  float* rs_o[9]; float* rs_i[9];
  for (int t = 0; t < 9; ++t) {
    rs_o[t] = ws + off; off += NNODES[SRC_T[t]];
    rs_i[t] = ws + off; off += NNODES[DST_T[t]];
  }
  long rs_total = off - rs_base;
  float* colsum = ws + off; off += D;               // colsum + colsq contiguous
  float* colsq  = ws + off; off += D;
  float* mu     = ws + off; off += D;
  float* inv    = ws + off; off += D;

  // ---- degrees -> rsqrt(max(deg,1)) (edges are layer-invariant) ----
  k_zero<<<512, 256, 0, stream>>>(ws + rs_base, rs_total);
  for (int t = 0; t < 9; ++t)
    k_deg<<<512, 256, 0, stream>>>(esrc[t], edst[t], ne[t], rs_o[t], rs_i[t]);
  k_rsqrt_deg<<<512, 256, 0, stream>>>(ws + rs_base, rs_total);

  // layer-0 accumulators = 0 ; layer-1 accumulators = d_out seeded with feats
  // (residual: final = feats + sum of layer-1 conv outputs)
  k_zero<<<4096, 256, 0, stream>>>(out0[0], (long)(NA_ + NB_ + NG_) * D);
  float* dout = (float*)d_out;
  float* out1[3] = {dout, dout + (long)NA_ * D, dout + (long)(NA_ + NB_) * D};
  k_copy<<<4096, 256, 0, stream>>>(out1[0], feat[0], (long)NA_ * D);
  k_copy<<<4096, 256, 0, stream>>>(out1[1], feat[1], (long)NB_ * D);
  k_copy<<<4096, 256, 0, stream>>>(out1[2], feat[2], (long)NG_ * D);

  for (int l = 0; l < 2; ++l) {
    const float* hin[3];
    float* hout[3];
    for (int n = 0; n < 3; ++n) {
      hin[n]  = l ? (const float*)out0[n] : feat[n];
      hout[n] = l ? out1[n] : out0[n];
    }
    for (int t = 0; t < 9; ++t) {
      const int st = SRC_T[t], dt = DST_T[t];
      const long ndst = NNODES[dt];

      k_zero<<<4096, 256, 0, stream>>>(agg, ndst * (long)D);
      k_scatter<<<4096, 256, 0, stream>>>(hin[st], esrc[t], edst[t], rs_o[t], agg, ne[t]);
      k_zero<<<4, 256, 0, stream>>>(colsum, 2 * (long)D);

      dim3 grid((unsigned)(ndst / BM), D / BN);     // ndst is a multiple of 16
      k_gemm_bn<<<grid, 256, 0, stream>>>(agg, rs_i[t], Wp[l][t], bp[l][t],
                                          colsum, colsq, mu, inv,
                                          gammap[l][t], betap[l][t], hout[dt], 1);
      k_bnstats<<<1, D, 0, stream>>>(colsum, colsq, mu, inv, (float)ndst);
      k_gemm_bn<<<grid, 256, 0, stream>>>(agg, rs_i[t], Wp[l][t], bp[l][t],
                                          colsum, colsq, mu, inv,
                                          gammap[l][t], betap[l][t], hout[dt], 2);
    }
  }
}